// DGCNN_66795331387612
// MI455X (gfx1250) — compile-verified
//
#include <hip/hip_runtime.h>
#include <hip/hip_bf16.h>

typedef float v2f __attribute__((ext_vector_type(2)));
typedef float v8f __attribute__((ext_vector_type(8)));

#define KNN 20
#define LEAK 0.2f
#define BN_EPS 1e-5f

constexpr int NPTS = 4096;   // N
constexpr int NBATCH = 2;    // B
constexpr int NLOG = 12;     // log2(NPTS)
constexpr int TILES_PER_B = NPTS / 16;  // 256

// ---------------------------------------------------------------------------
// transpose pts (B,2,N) -> pts_t (B,N,2)
// ---------------------------------------------------------------------------
__global__ void dgcnn_transpose_pts(const float* __restrict__ pts,
                                    float* __restrict__ pts_t) {
  int i = blockIdx.x * blockDim.x + threadIdx.x;  // over B*2*N
  if (i >= NBATCH * 2 * NPTS) return;
  int n = i & (NPTS - 1);
  int c = (i >> NLOG) & 1;
  int b = i >> (NLOG + 1);
  pts_t[((size_t)(b * NPTS + n)) * 2 + c] = pts[i];
}

__global__ void dgcnn_zero_f32(float* __restrict__ p, int count) {
  int i = blockIdx.x * blockDim.x + threadIdx.x;
  if (i < count) p[i] = 0.0f;
}

// ---------------------------------------------------------------------------
// knn: one wave per query point. Per-lane register top-K (fully unrolled
// insertion => stays in VGPRs), then wave-wide shuffle merge.
// ---------------------------------------------------------------------------
__global__ __launch_bounds__(256) void dgcnn_knn(
    const float* __restrict__ pts_t, int* __restrict__ idx_out) {
  int gw = (blockIdx.x * 256 + threadIdx.x) >> 5;  // global wave id = query
  int lane = threadIdx.x & 31;
  if (gw >= NBATCH * NPTS) return;
  int b = gw >> NLOG, n = gw & (NPTS - 1);

  float qx = pts_t[(size_t)(b * NPTS + n) * 2 + 0];
  float qy = pts_t[(size_t)(b * NPTS + n) * 2 + 1];

  float d[KNN];
  int id[KNN];
#pragma unroll
  for (int j = 0; j < KNN; ++j) { d[j] = 3.4e38f; id[j] = -1; }

  for (int m = lane; m < NPTS; m += 32) {
    float px = pts_t[(size_t)(b * NPTS + m) * 2 + 0];
    float py = pts_t[(size_t)(b * NPTS + m) * 2 + 1];
    float dx = qx - px, dy = qy - py;
    float dist = dx * dx + dy * dy;
    if (dist < d[KNN - 1]) {
      d[KNN - 1] = dist; id[KNN - 1] = m;
#pragma unroll
      for (int j = KNN - 1; j > 0; --j) {  // one bubble pass restores order
        if (d[j] < d[j - 1]) {
          float td = d[j]; d[j] = d[j - 1]; d[j - 1] = td;
          int ti = id[j]; id[j] = id[j - 1]; id[j - 1] = ti;
        }
      }
    }
  }

  // merge the 32 sorted per-lane lists, K rounds of wave-argmin
  int ptr = 0;
  for (int r = 0; r < KNN; ++r) {
    float curD = 3.4e38f; int curI = -1;
#pragma unroll
    for (int j = 0; j < KNN; ++j)
      if (j == ptr) { curD = d[j]; curI = id[j]; }
    int owner = lane;
    for (int off = 16; off; off >>= 1) {
      float od = __shfl_xor(curD, off, 32);
      int oi = __shfl_xor(curI, off, 32);
      int oo = __shfl_xor(owner, off, 32);
      if (od < curD || (od == curD && oo < owner)) { curD = od; curI = oi; owner = oo; }
    }
    if (lane == owner) ptr++;
    if (lane == 0) idx_out[(size_t)gw * KNN + r] = curI;
  }
}

// ---------------------------------------------------------------------------
// Fused EdgeConv: LDS-staged gather -> fully unrolled WMMA f32 16x16x4 GEMM
// -> running max/min over K + per-channel sum/sumsq for BN stats.
// CIN/OC are compile-time: k-step loop fully unrolls into wmma clauses.
// ---------------------------------------------------------------------------
template <int CIN, int OC>
__global__ __launch_bounds__(256) void dgcnn_edge_conv_wmma(
    const float* __restrict__ xt, int rowStride, int inOff,
    const int* __restrict__ idx, const float* __restrict__ w,  // OC x 2CIN
    float* __restrict__ maxbuf, float* __restrict__ minbuf,    // B*OC*N
    float* __restrict__ sums, float* __restrict__ sumsq) {
  constexpr int C2 = 2 * CIN;
  constexpr int NSTEPS = C2 / 4;          // WMMA k-steps
  constexpr int NOT = OC / 16;            // o-tiles
  constexpr int SLOTS = (NOT + 7) / 8;

  __shared__ float sC[CIN * 16];   // center features  [c][n]
  __shared__ float sNb[CIN * 16];  // neighbor features for current k
  __shared__ int sIdx[16];

  const int tid = threadIdx.x;
  const int lane = tid & 31;
  const int wid = tid >> 5;                       // 0..7
  const int b = blockIdx.x >> 8;                  // tile / 256
  const int n0 = (blockIdx.x & 255) * 16;

  // prefetch weight rows into cache (global_prefetch_b8)
  for (int r = tid; r < OC; r += 256)
    __builtin_prefetch(&w[(size_t)r * C2], 0, 3);

  // stage center features (k-invariant)
  if constexpr (CIN >= 4) {
    constexpr int C4 = CIN / 4;
    for (int e = tid; e < C4 * 16; e += 256) {
      int nl = e / C4, c4 = e - nl * C4;
      const float4 vv = *(const float4*)&xt[(size_t)(b * NPTS + n0 + nl) * rowStride + inOff + c4 * 4];
      sC[(c4 * 4 + 0) * 16 + nl] = vv.x;
      sC[(c4 * 4 + 1) * 16 + nl] = vv.y;
      sC[(c4 * 4 + 2) * 16 + nl] = vv.z;
      sC[(c4 * 4 + 3) * 16 + nl] = vv.w;
    }
  } else {
    for (int e = tid; e < CIN * 16; e += 256) {
      int nl = e / CIN, c = e - nl * CIN;
      sC[c * 16 + nl] = xt[(size_t)(b * NPTS + n0 + nl) * rowStride + inOff + c];
    }
  }

  float mx[SLOTS][8], mn[SLOTS][8], sm[SLOTS][8], sq[SLOTS][8];
#pragma unroll
  for (int t = 0; t < SLOTS; ++t)
#pragma unroll
    for (int v = 0; v < 8; ++v) {
      mx[t][v] = -3.4e38f; mn[t][v] = 3.4e38f; sm[t][v] = 0.f; sq[t][v] = 0.f;
    }

  const int col = lane & 15;            // B col / A row m / D col
  const int kb = (lane < 16) ? 0 : 2;   // K sub-offset per ISA f32 layout

  for (int k = 0; k < KNN; ++k) {
    __syncthreads();
    if (tid < 16) sIdx[tid] = idx[((size_t)(b * NPTS + n0 + tid)) * KNN + k];
    __syncthreads();
    if constexpr (CIN >= 4) {
      constexpr int C4 = CIN / 4;
      for (int e = tid; e < C4 * 16; e += 256) {
        int nl = e / C4, c4 = e - nl * C4;
        const float4 vv = *(const float4*)&xt[(size_t)(b * NPTS + sIdx[nl]) * rowStride + inOff + c4 * 4];
        sNb[(c4 * 4 + 0) * 16 + nl] = vv.x;
        sNb[(c4 * 4 + 1) * 16 + nl] = vv.y;
        sNb[(c4 * 4 + 2) * 16 + nl] = vv.z;
        sNb[(c4 * 4 + 3) * 16 + nl] = vv.w;
      }
    } else {
      for (int e = tid; e < CIN * 16; e += 256) {
        int nl = e / CIN, c = e - nl * CIN;
        sNb[c * 16 + nl] = xt[(size_t)(b * NPTS + sIdx[nl]) * rowStride + inOff + c];
      }
    }
    __syncthreads();

    int slot = 0;
    for (int ot = wid; ot < NOT; ot += 8, ++slot) {
      int o0 = ot * 16;
      v8f acc = {0.f, 0.f, 0.f, 0.f, 0.f, 0.f, 0.f, 0.f};
#pragma unroll
      for (int cs = 0; cs < NSTEPS; ++cs) {
        int c0 = cs * 4 + kb;
        v2f a;
        a.x = w[(size_t)(o0 + col) * C2 + c0];
        a.y = w[(size_t)(o0 + col) * C2 + c0 + 1];
        v2f bb;
        bb.x = (c0 < CIN) ? (sNb[c0 * 16 + col] - sC[c0 * 16 + col])
                          : sC[(c0 - CIN) * 16 + col];
        int c1 = c0 + 1;
        bb.y = (c1 < CIN) ? (sNb[c1 * 16 + col] - sC[c1 * 16 + col])
                          : sC[(c1 - CIN) * 16 + col];
        acc = __builtin_amdgcn_wmma_f32_16x16x4_f32(false, a, false, bb,
                                                    (short)0, acc, false, false);
      }
#pragma unroll
      for (int v = 0; v < 8; ++v) {
        float y = acc[v];
        mx[slot][v] = fmaxf(mx[slot][v], y);
        mn[slot][v] = fminf(mn[slot][v], y);
        sm[slot][v] += y;
        sq[slot][v] += y * y;
      }
    }
  }

  // writeback: D layout -> lane l holds (row v + 8*(l/16), col l%16)
  int rowBase = (lane >= 16) ? 8 : 0;
  int slot = 0;
  for (int ot = wid; ot < NOT; ot += 8, ++slot) {
    int o0 = ot * 16;
#pragma unroll
    for (int v = 0; v < 8; ++v) {
      int o = o0 + rowBase + v;
      size_t off = ((size_t)(b * OC + o)) * NPTS + n0 + col;
      maxbuf[off] = mx[slot][v];
      minbuf[off] = mn[slot][v];
    }
#pragma unroll
    for (int v = 0; v < 8; ++v) {
      float s = sm[slot][v], s2 = sq[slot][v];
#pragma unroll
      for (int dlt = 1; dlt < 16; dlt <<= 1) {  // reduce over 16 columns
        s += __shfl_xor(s, dlt, 32);
        s2 += __shfl_xor(s2, dlt, 32);
      }
      if (col == 0) {
        int o = o0 + rowBase + v;
        atomicAdd(&sums[o], s);
        atomicAdd(&sumsq[o], s2);
      }
    }
  }
}

// ---------------------------------------------------------------------------
// BN stats finalize: mean and a = g * rsqrt(var + eps)
// ---------------------------------------------------------------------------
__global__ void dgcnn_finalize_stats(const float* __restrict__ sums,
                                     const float* __restrict__ sumsq,
                                     const float* __restrict__ g, int O,
                                     float invCount, float* __restrict__ mean,
                                     float* __restrict__ ascale) {
  int o = blockIdx.x * blockDim.x + threadIdx.x;
  if (o >= O) return;
  float mu = sums[o] * invCount;
  float var = sumsq[o] * invCount - mu * mu;
  mean[o] = mu;
  ascale[o] = g[o] * rsqrtf(var + BN_EPS);
}

// ---------------------------------------------------------------------------
// apply BN+LReLU after max-over-K. max/affine commute only for a>=0, so pick
// max or min by sign(a). Writes transposed into the concat buffer slice.
// ---------------------------------------------------------------------------
template <int OC>
__global__ void dgcnn_apply_edge(const float* __restrict__ maxbuf,
                                 const float* __restrict__ minbuf,
                                 const float* __restrict__ mean,
                                 const float* __restrict__ ascale,
                                 const float* __restrict__ bias,
                                 float* __restrict__ xcat_t, int outOff) {
  int i = blockIdx.x * blockDim.x + threadIdx.x;  // over B*OC*N
  if (i >= NBATCH * OC * NPTS) return;
  int n = i & (NPTS - 1);
  int o = (i >> NLOG) & (OC - 1);
  int b = i / (NPTS * OC);
  float a = ascale[o];
  float v = (a >= 0.f) ? maxbuf[i] : minbuf[i];
  float y = (v - mean[o]) * a + bias[o];
  y = (y >= 0.f) ? y : LEAK * y;
  xcat_t[((size_t)(b * NPTS + n)) * 512 + outOff + o] = y;
}

// ---------------------------------------------------------------------------
// Final 512x512 conv1d over concat features (B,N,512) -> raw y + BN stats
// ---------------------------------------------------------------------------
__global__ __launch_bounds__(256) void dgcnn_final_conv_wmma(
    const float* __restrict__ xt,   // (B*N, 512)
    const float* __restrict__ w,    // 512 x 512
    float* __restrict__ ybuf,       // B*512*N raw pre-BN
    float* __restrict__ sums, float* __restrict__ sumsq) {
  __shared__ float sX[512 * 16];
  const int tid = threadIdx.x;
  const int lane = tid & 31;
  const int wid = tid >> 5;
  const int b = blockIdx.x >> 8;
  const int n0 = (blockIdx.x & 255) * 16;

  for (int e = tid; e < 128 * 16; e += 256) {
    int nl = e >> 7, c4 = e & 127;
    const float4 vv = *(const float4*)&xt[(size_t)(b * NPTS + n0 + nl) * 512 + c4 * 4];
    sX[(c4 * 4 + 0) * 16 + nl] = vv.x;
    sX[(c4 * 4 + 1) * 16 + nl] = vv.y;
    sX[(c4 * 4 + 2) * 16 + nl] = vv.z;
    sX[(c4 * 4 + 3) * 16 + nl] = vv.w;
  }
  __syncthreads();

  const int col = lane & 15;
  const int kb = (lane < 16) ? 0 : 2;
  const int rowBase = (lane >= 16) ? 8 : 0;

  for (int ot = wid; ot < 32; ot += 8) {
    int o0 = ot * 16;
    v8f acc = {0.f, 0.f, 0.f, 0.f, 0.f, 0.f, 0.f, 0.f};
#pragma unroll 8
    for (int cs = 0; cs < 128; ++cs) {
      int c0 = cs * 4 + kb;
      v2f a;
      a.x = w[(size_t)(o0 + col) * 512 + c0];
      a.y = w[(size_t)(o0 + col) * 512 + c0 + 1];
      v2f bb;
      bb.x = sX[c0 * 16 + col];
      bb.y = sX[(c0 + 1) * 16 + col];
      acc = __builtin_amdgcn_wmma_f32_16x16x4_f32(false, a, false, bb, (short)0,
                                                  acc, false, false);
    }
#pragma unroll
    for (int v = 0; v < 8; ++v) {
      int o = o0 + rowBase + v;
      ybuf[((size_t)(b * 512 + o)) * NPTS + n0 + col] = acc[v];
    }
#pragma unroll
    for (int v = 0; v < 8; ++v) {
      float s = acc[v], s2 = acc[v] * acc[v];
#pragma unroll
      for (int dlt = 1; dlt < 16; dlt <<= 1) {
        s += __shfl_xor(s, dlt, 32);
        s2 += __shfl_xor(s2, dlt, 32);
      }
      if (col == 0) {
        int o = o0 + rowBase + v;
        atomicAdd(&sums[o], s);
        atomicAdd(&sumsq[o], s2);
      }
    }
  }
}

__global__ void dgcnn_apply_final(const float* __restrict__ ybuf,
                                  const float* __restrict__ mean,
                                  const float* __restrict__ ascale,
                                  const float* __restrict__ bias,
                                  float* __restrict__ out) {
  int i = blockIdx.x * blockDim.x + threadIdx.x;  // over B*512*N
  if (i >= NBATCH * 512 * NPTS) return;
  int o = (i >> NLOG) & 511;
  float y = (ybuf[i] - mean[o]) * ascale[o] + bias[o];
  out[i] = (y >= 0.f) ? y : LEAK * y;
}

// ---------------------------------------------------------------------------
extern "C" void kernel_launch(void* const* d_in, const int* in_sizes, int n_in,
                              void* d_out, int out_size, void* d_ws,
                              size_t ws_size, hipStream_t stream) {
  const float* pts = (const float*)d_in[0];
  const float* W[5] = {(const float*)d_in[1], (const float*)d_in[4],
                       (const float*)d_in[7], (const float*)d_in[10],
                       (const float*)d_in[13]};
  const float* G[5] = {(const float*)d_in[2], (const float*)d_in[5],
                       (const float*)d_in[8], (const float*)d_in[11],
                       (const float*)d_in[14]};
  const float* Bv[5] = {(const float*)d_in[3], (const float*)d_in[6],
                        (const float*)d_in[9], (const float*)d_in[12],
                        (const float*)d_in[15]};

  // workspace carve-up (256B aligned)
  char* ws = (char*)d_ws;
  size_t off = 0;
  auto carve = [&](size_t bytes) {
    size_t cur = off;
    off += (bytes + 255) & ~(size_t)255;
    return cur;
  };
  int* idx = (int*)(ws + carve((size_t)NBATCH * NPTS * KNN * sizeof(int)));
  float* pts_t = (float*)(ws + carve((size_t)NBATCH * NPTS * 2 * sizeof(float)));
  float* xcat_t = (float*)(ws + carve((size_t)NBATCH * NPTS * 512 * sizeof(float)));
  float* maxbuf = (float*)(ws + carve((size_t)NBATCH * 256 * NPTS * sizeof(float)));
  float* minbuf = (float*)(ws + carve((size_t)NBATCH * 256 * NPTS * sizeof(float)));
  float* ybuf = (float*)(ws + carve((size_t)NBATCH * 512 * NPTS * sizeof(float)));
  float* stats = (float*)(ws + carve(4 * 512 * sizeof(float)));
  float* sums = stats;
  float* sq = stats + 512;
  float* mean = stats + 1024;
  float* asc = stats + 1536;

  dgcnn_transpose_pts<<<(NBATCH * 2 * NPTS + 255) / 256, 256, 0, stream>>>(pts, pts_t);
  dgcnn_knn<<<(NBATCH * NPTS * 32) / 256, 256, 0, stream>>>(pts_t, idx);

  const int nTiles = NBATCH * TILES_PER_B;  // 512 workgroups

  // ---- edge block 1: Cin=2, O=64, in=pts_t, out channels [0,64) ----
  dgcnn_zero_f32<<<4, 256, 0, stream>>>(stats, 1024);
  dgcnn_edge_conv_wmma<2, 64><<<nTiles, 256, 0, stream>>>(
      pts_t, 2, 0, idx, W[0], maxbuf, minbuf, sums, sq);
  dgcnn_finalize_stats<<<1, 128, 0, stream>>>(
      sums, sq, G[0], 64, 1.0f / (float)(NBATCH * NPTS * KNN), mean, asc);
  dgcnn_apply_edge<64><<<(NBATCH * 64 * NPTS + 255) / 256, 256, 0, stream>>>(
      maxbuf, minbuf, mean, asc, Bv[0], xcat_t, 0);

  // ---- edge block 2: Cin=64, O=64, in=[0,64), out [64,128) ----
  dgcnn_zero_f32<<<4, 256, 0, stream>>>(stats, 1024);
  dgcnn_edge_conv_wmma<64, 64><<<nTiles, 256, 0, stream>>>(
      xcat_t, 512, 0, idx, W[1], maxbuf, minbuf, sums, sq);
  dgcnn_finalize_stats<<<1, 128, 0, stream>>>(
      sums, sq, G[1], 64, 1.0f / (float)(NBATCH * NPTS * KNN), mean, asc);
  dgcnn_apply_edge<64><<<(NBATCH * 64 * NPTS + 255) / 256, 256, 0, stream>>>(
      maxbuf, minbuf, mean, asc, Bv[1], xcat_t, 64);

  // ---- edge block 3: Cin=64, O=128, in=[64,128), out [128,256) ----
  dgcnn_zero_f32<<<4, 256, 0, stream>>>(stats, 1024);
  dgcnn_edge_conv_wmma<64, 128><<<nTiles, 256, 0, stream>>>(
      xcat_t, 512, 64, idx, W[2], maxbuf, minbuf, sums, sq);
  dgcnn_finalize_stats<<<1, 128, 0, stream>>>(
      sums, sq, G[2], 128, 1.0f / (float)(NBATCH * NPTS * KNN), mean, asc);
  dgcnn_apply_edge<128><<<(NBATCH * 128 * NPTS + 255) / 256, 256, 0, stream>>>(
      maxbuf, minbuf, mean, asc, Bv[2], xcat_t, 128);

  // ---- edge block 4: Cin=128, O=256, in=[128,256), out [256,512) ----
  dgcnn_zero_f32<<<4, 256, 0, stream>>>(stats, 1024);
  dgcnn_edge_conv_wmma<128, 256><<<nTiles, 256, 0, stream>>>(
      xcat_t, 512, 128, idx, W[3], maxbuf, minbuf, sums, sq);
  dgcnn_finalize_stats<<<2, 128, 0, stream>>>(
      sums, sq, G[3], 256, 1.0f / (float)(NBATCH * NPTS * KNN), mean, asc);
  dgcnn_apply_edge<256><<<(NBATCH * 256 * NPTS + 255) / 256, 256, 0, stream>>>(
      maxbuf, minbuf, mean, asc, Bv[3], xcat_t, 256);

  // ---- final 512x512 conv + BN + LReLU ----
  dgcnn_zero_f32<<<4, 256, 0, stream>>>(stats, 1024);
  dgcnn_final_conv_wmma<<<nTiles, 256, 0, stream>>>(xcat_t, W[4], ybuf, sums, sq);
  dgcnn_finalize_stats<<<4, 128, 0, stream>>>(
      sums, sq, G[4], 512, 1.0f / (float)(NBATCH * NPTS), mean, asc);
  dgcnn_apply_final<<<(NBATCH * 512 * NPTS + 255) / 256, 256, 0, stream>>>(
      ybuf, mean, asc, Bv[4], (float*)d_out);
}